// ChebNet_15530601743030
// MI455X (gfx1250) — compile-verified
//
#include <hip/hip_runtime.h>
#include <math.h>

#define N_NODES 100000   // divisible by 16 -> no row guards needed in WMMA tiles
#define N_EDGES 1600000
#define F_IN    50
#define HID     32
#define NCLS    40

typedef float v2f __attribute__((ext_vector_type(2)));
typedef float v8f __attribute__((ext_vector_type(8)));

// ---------------------------------------------------------------- utilities

__global__ void zero_kernel(float* __restrict__ p, long long n) {
  long long i = blockIdx.x * (long long)blockDim.x + threadIdx.x;
  if (i < n) p[i] = 0.0f;
}

// Wcat1: 52 x 64, rows k<50 = [W0_1 | W1_1], rest zero padded.
__global__ void pack_w1_kernel(const float* __restrict__ W0,
                               const float* __restrict__ W1,
                               float* __restrict__ Wcat) {
  int t = blockIdx.x * blockDim.x + threadIdx.x;
  if (t >= 52 * 64) return;
  int k = t >> 6, j = t & 63;
  float v = 0.0f;
  if (k < F_IN) v = (j < HID) ? W0[k * HID + j] : W1[k * HID + (j - HID)];
  Wcat[t] = v;
}

// Wcat2: 64 x 48, rows 0..31 = W0_2, rows 32..63 = W1_2, cols 40..47 zero.
__global__ void pack_w2_kernel(const float* __restrict__ W0,
                               const float* __restrict__ W1,
                               float* __restrict__ Wcat) {
  int t = blockIdx.x * blockDim.x + threadIdx.x;
  if (t >= 64 * 48) return;
  int k = t / 48, j = t % 48;
  float v = 0.0f;
  if (j < NCLS) v = (k < HID) ? W0[k * NCLS + j] : W1[(k - HID) * NCLS + j];
  Wcat[t] = v;
}

// ---------------------------------------------------------------- graph norm

__global__ void deg_kernel(const int* __restrict__ rows,
                           const int* __restrict__ cols,
                           float* __restrict__ deg) {
  int e = blockIdx.x * blockDim.x + threadIdx.x;
  if (e >= N_EDGES) return;
  int r = rows[e], c = cols[e];
  if (r != c) atomicAdd(&deg[r], 1.0f);
}

__global__ void dis_kernel(const float* __restrict__ deg, float* __restrict__ dis) {
  int i = blockIdx.x * blockDim.x + threadIdx.x;
  if (i >= N_NODES) return;
  float d = deg[i];
  dis[i] = (d > 0.0f) ? rsqrtf(fmaxf(d, 1.0f)) : 0.0f;
}

// ------------------------------------------------ edge scatter (segment_sum)
// dst[r*32 + f] += (-dis[r]*dis[c]) * src[c*sstride + soff + f], f = 0..31
// One wave per edge (lane = feature): edge metadata loads are wave-uniform,
// the 32-float gather is one coalesced cacheline, scatter is f32 atomics.

__global__ void agg_kernel(const int* __restrict__ rows,
                           const int* __restrict__ cols,
                           const float* __restrict__ dis,
                           const float* __restrict__ src, int sstride, int soff,
                           float* __restrict__ dst) {
  long long gid = blockIdx.x * (long long)blockDim.x + threadIdx.x;
  long long e = gid >> 5;
  if (e >= (long long)N_EDGES) return;
  int f = (int)(gid & 31);
  int r = rows[e], c = cols[e];
  if (r == c) return;
  float w = -dis[r] * dis[c];
  if (w == 0.0f) return;
  atomicAdd(&dst[(long long)r * HID + f],
            w * src[(long long)c * sstride + soff + f]);
}

// --------------------------------------------- WMMA GEMM 1: P = x @ Wcat1
// x: N x 50 (row major), Wcat1: 52 x 64 (zero padded), P: N x 64.
// Fragment layouts per CDNA5 ISA 7.12.2 (wave32, f32 16x16x4):
//   A 16x4 : lane<16 -> M=lane,   VGPR{0,1} = K{kk,kk+1};  lane>=16 -> K{kk+2,kk+3}
//   B 4x16 : N = lane&15, K = kk + vgpr + 2*(lane>=16)
//   C 16x16: M = vgpr + 8*(lane>=16), N = lane&15
// No row guards: N_NODES % 16 == 0. K tail handled branch-free (cndmask).

__global__ void __launch_bounds__(128)
proj1_wmma_kernel(const float* __restrict__ x,
                  const float* __restrict__ Wcat,   // 52 x 64
                  float* __restrict__ P) {          // N x 64
  int wave = threadIdx.x >> 5;
  int lane = threadIdx.x & 31;
  int half = lane >> 4;
  int l16  = lane & 15;
  int m0 = (blockIdx.x * 4 + wave) * 16;
  if (m0 >= N_NODES) return;                 // wave-uniform: EXEC stays full
  int mA = m0 + l16;
  const float* arow = x + (long long)mA * F_IN;

  v8f acc0 = {}, acc1 = {}, acc2 = {}, acc3 = {};
  // main steps: k = kk + 2*half <= 46, k+1 <= 47 < 50 -> unconditional
  for (int kk = 0; kk < 48; kk += 4) {
    int k = kk + 2 * half;
    v2f a = *(const v2f*)(arow + k);         // 8B-aligned global_load_b64
    const float* b0 = Wcat + k * 64;
    const float* b1 = Wcat + (k + 1) * 64;
    v2f b;
    b.x = b0[l16];      b.y = b1[l16];
    acc0 = __builtin_amdgcn_wmma_f32_16x16x4_f32(false, a, false, b, (short)0, acc0, false, false);
    b.x = b0[16 + l16]; b.y = b1[16 + l16];
    acc1 = __builtin_amdgcn_wmma_f32_16x16x4_f32(false, a, false, b, (short)0, acc1, false, false);
    b.x = b0[32 + l16]; b.y = b1[32 + l16];
    acc2 = __builtin_amdgcn_wmma_f32_16x16x4_f32(false, a, false, b, (short)0, acc2, false, false);
    b.x = b0[48 + l16]; b.y = b1[48 + l16];
    acc3 = __builtin_amdgcn_wmma_f32_16x16x4_f32(false, a, false, b, (short)0, acc3, false, false);
  }
  // epilogue: kk=48 -> K = {48,49,50,51}; upper half-wave (k=50,51) is zero pad
  {
    int k = 48 + 2 * half;
    v2f aload = *(const v2f*)(arow + 48);    // always in-bounds
    v2f a;
    a.x = half ? 0.0f : aload.x;             // v_cndmask, no EXEC branches
    a.y = half ? 0.0f : aload.y;
    const float* b0 = Wcat + k * 64;         // rows 50,51 are zero padded
    const float* b1 = Wcat + (k + 1) * 64;
    v2f b;
    b.x = b0[l16];      b.y = b1[l16];
    acc0 = __builtin_amdgcn_wmma_f32_16x16x4_f32(false, a, false, b, (short)0, acc0, false, false);
    b.x = b0[16 + l16]; b.y = b1[16 + l16];
    acc1 = __builtin_amdgcn_wmma_f32_16x16x4_f32(false, a, false, b, (short)0, acc1, false, false);
    b.x = b0[32 + l16]; b.y = b1[32 + l16];
    acc2 = __builtin_amdgcn_wmma_f32_16x16x4_f32(false, a, false, b, (short)0, acc2, false, false);
    b.x = b0[48 + l16]; b.y = b1[48 + l16];
    acc3 = __builtin_amdgcn_wmma_f32_16x16x4_f32(false, a, false, b, (short)0, acc3, false, false);
  }
#pragma unroll
  for (int v = 0; v < 8; v++) {
    int mC = m0 + v + 8 * half;              // always < N_NODES
    float* prow = P + (long long)mC * 64 + l16;
    prow[0]  = acc0[v];
    prow[16] = acc1[v];
    prow[32] = acc2[v];
    prow[48] = acc3[v];
  }
}

// ---------------------------------------- h = relu(P[:, :32] + agg1 + b1)

__global__ void relu_kernel(const float* __restrict__ P,
                            const float* __restrict__ agg1,
                            const float* __restrict__ b1,
                            float* __restrict__ h) {
  long long i = blockIdx.x * (long long)blockDim.x + threadIdx.x;
  if (i >= (long long)N_NODES * HID) return;
  long long n = i >> 5;
  int f = (int)(i & 31);
  float v = P[n * 64 + f] + agg1[i] + b1[f];
  h[i] = v > 0.0f ? v : 0.0f;
}

// ----------------------- WMMA GEMM 2: outp = [h | agg2] @ Wcat2  (K = 64)

__global__ void __launch_bounds__(128)
out_wmma_kernel(const float* __restrict__ h,     // N x 32
                const float* __restrict__ agg2,  // N x 32
                const float* __restrict__ Wcat,  // 64 x 48
                float* __restrict__ outp) {      // N x 48
  int wave = threadIdx.x >> 5;
  int lane = threadIdx.x & 31;
  int half = lane >> 4;
  int l16  = lane & 15;
  int m0 = (blockIdx.x * 4 + wave) * 16;
  if (m0 >= N_NODES) return;
  int mA = m0 + l16;
  const float* hrow = h    + (long long)mA * HID;
  const float* grow = agg2 + (long long)mA * HID - HID;  // index with k directly

  v8f acc0 = {}, acc1 = {}, acc2 = {};
  for (int kk = 0; kk < 64; kk += 4) {
    int k = kk + 2 * half;   // even; (k, k+1) never straddles the 32 boundary
    const float* src = (k < HID) ? (hrow + k) : (grow + k);  // cndmask select
    v2f a = *(const v2f*)src;                // 8B-aligned global_load_b64
    const float* b0 = Wcat + k * 48;
    const float* b1 = Wcat + (k + 1) * 48;
    v2f b;
    b.x = b0[l16];      b.y = b1[l16];
    acc0 = __builtin_amdgcn_wmma_f32_16x16x4_f32(false, a, false, b, (short)0, acc0, false, false);
    b.x = b0[16 + l16]; b.y = b1[16 + l16];
    acc1 = __builtin_amdgcn_wmma_f32_16x16x4_f32(false, a, false, b, (short)0, acc1, false, false);
    b.x = b0[32 + l16]; b.y = b1[32 + l16];
    acc2 = __builtin_amdgcn_wmma_f32_16x16x4_f32(false, a, false, b, (short)0, acc2, false, false);
  }
#pragma unroll
  for (int v = 0; v < 8; v++) {
    int mC = m0 + v + 8 * half;              // always < N_NODES
    float* prow = outp + (long long)mC * 48 + l16;
    prow[0]  = acc0[v];
    prow[16] = acc1[v];
    prow[32] = acc2[v];
  }
}

// --------------------------------------------------- per-row log_softmax

__global__ void logsoftmax_kernel(const float* __restrict__ outp,  // N x 48
                                  const float* __restrict__ b2,
                                  float* __restrict__ out) {       // N x 40
  int i = blockIdx.x * blockDim.x + threadIdx.x;
  if (i >= N_NODES) return;
  const float* r = outp + (long long)i * 48;
  float m = -INFINITY;
  for (int j = 0; j < NCLS; j++) m = fmaxf(m, r[j] + b2[j]);
  float s = 0.0f;
  for (int j = 0; j < NCLS; j++) s += expf(r[j] + b2[j] - m);
  float ls = logf(s);
  for (int j = 0; j < NCLS; j++)
    out[(long long)i * NCLS + j] = (r[j] + b2[j] - m) - ls;
}

// ---------------------------------------------------------------- launcher

extern "C" void kernel_launch(void* const* d_in, const int* in_sizes, int n_in,
                              void* d_out, int out_size, void* d_ws, size_t ws_size,
                              hipStream_t stream) {
  (void)in_sizes; (void)n_in; (void)out_size; (void)ws_size;
  const float* x    = (const float*)d_in[0];
  const int*   ei   = (const int*)  d_in[1];
  const float* W0_1 = (const float*)d_in[2];
  const float* W1_1 = (const float*)d_in[3];
  const float* b1   = (const float*)d_in[4];
  const float* W0_2 = (const float*)d_in[5];
  const float* W1_2 = (const float*)d_in[6];
  const float* b2   = (const float*)d_in[7];
  float* out = (float*)d_out;

  const int* rows = ei;
  const int* cols = ei + N_EDGES;

  const long long N = N_NODES;
  float* ws   = (float*)d_ws;
  float* deg  = ws;              // N
  float* dis  = ws + N;          // N
  float* P    = ws + 2 * N;      // N x 64; later reused as outp (N x 48)
  float* agg1 = ws + 66 * N;     // N x 32; later reused as agg2
  float* hbuf = ws + 98 * N;     // N x 32
  float* Wc1  = ws + 130 * N;    // 52 x 64
  float* Wc2  = Wc1 + 52 * 64;   // 64 x 48
  // total: 130*N + 6400 floats ~= 52 MB

  const int T = 256;
  // init / pack
  zero_kernel<<<(unsigned)((N + T - 1) / T), T, 0, stream>>>(deg, N);
  zero_kernel<<<(unsigned)((32 * N + T - 1) / T), T, 0, stream>>>(agg1, 32 * N);
  pack_w1_kernel<<<(52 * 64 + T - 1) / T, T, 0, stream>>>(W0_1, W1_1, Wc1);
  pack_w2_kernel<<<(64 * 48 + T - 1) / T, T, 0, stream>>>(W0_2, W1_2, Wc2);

  // symmetric normalization
  deg_kernel<<<(N_EDGES + T - 1) / T, T, 0, stream>>>(rows, cols, deg);
  dis_kernel<<<(unsigned)((N + T - 1) / T), T, 0, stream>>>(deg, dis);

  // layer 1: project (WMMA) -> aggregate -> relu+bias
  proj1_wmma_kernel<<<(N_NODES + 63) / 64, 128, 0, stream>>>(x, Wc1, P);
  agg_kernel<<<(unsigned)(((long long)N_EDGES * 32 + T - 1) / T), T, 0, stream>>>(
      rows, cols, dis, P, 64, 32, agg1);
  relu_kernel<<<(unsigned)((32 * N + T - 1) / T), T, 0, stream>>>(P, agg1, b1, hbuf);

  // layer 2: aggregate h -> combined GEMM (WMMA) -> log_softmax
  zero_kernel<<<(unsigned)((32 * N + T - 1) / T), T, 0, stream>>>(agg1, 32 * N);
  agg_kernel<<<(unsigned)(((long long)N_EDGES * 32 + T - 1) / T), T, 0, stream>>>(
      rows, cols, dis, hbuf, 32, 0, agg1);
  out_wmma_kernel<<<(N_NODES + 63) / 64, 128, 0, stream>>>(hbuf, agg1, Wc2, P);
  logsoftmax_kernel<<<(unsigned)((N + T - 1) / T), T, 0, stream>>>(P, b2, out);
}